// ProteinDNAGNN_mini_22076131901586
// MI455X (gfx1250) — compile-verified
//
#include <hip/hip_runtime.h>
#include <hip/hip_bf16.h>
#include <math.h>

#define HID 128
#define NUM_GRAPHS 64
#define EPS 1e-5f

typedef float v2f __attribute__((ext_vector_type(2)));
typedef float v8f __attribute__((ext_vector_type(8)));

// ---------------------------------------------------------------------------
// small utility kernels
// ---------------------------------------------------------------------------
__global__ void zero_f32(float* __restrict__ p, int n) {
    int i = blockIdx.x * blockDim.x + threadIdx.x;
    if (i < n) p[i] = 0.0f;
}

__global__ void fill_neginf(float* __restrict__ p, int n) {
    int i = blockIdx.x * blockDim.x + threadIdx.x;
    if (i < n) p[i] = -INFINITY;
}

// deg accumulation: dis[dst] += 1  (dis buffer doubles as deg buffer)
__global__ void deg_accum(const int* __restrict__ dst, float* __restrict__ deg, int E) {
    int e = blockIdx.x * blockDim.x + threadIdx.x;
    if (e < E) unsafeAtomicAdd(&deg[dst[e]], 1.0f);
}

// dis = rsqrt(deg + 1)
__global__ void deg_to_dis(float* __restrict__ dis, int n) {
    int i = blockIdx.x * blockDim.x + threadIdx.x;
    if (i < n) dis[i] = rsqrtf(dis[i] + 1.0f);
}

// ---------------------------------------------------------------------------
// WMMA fp32 GEMM:  H[N,128] = X[N,128] @ W[128,128]
// One block = 256 threads = 8 waves; each wave owns a 16-row M-block and
// computes the full 16x128 output strip as 8 C-tiles of v_wmma_f32_16x16x4_f32.
//
// W is staged in LDS *pair-packed along K*: element (k,n) lives at
//   sW[(k>>1)*256 + n*2 + (k&1)]
// so a B fragment (rows kb,kb+1 at column n) is ONE contiguous 8-byte
// ds_load_b64 into an even-aligned VGPR pair — no re-pairing movs.
// ---------------------------------------------------------------------------
__global__ void gemm_wmma_f32(const float* __restrict__ X,
                              const float* __restrict__ W,
                              float* __restrict__ H, int N) {
    __shared__ float sW[HID * HID];   // 64 KB (<= 320 KB/WGP)

    const int tid = threadIdx.x;
    // cooperative load of W, scatter into pair-packed LDS layout
    for (int i = tid * 4; i < HID * HID; i += blockDim.x * 4) {
        const int k = i >> 7;          // row of W
        const int n = i & 127;         // col start (multiple of 4)
        float4 w = *(const float4*)(W + i);
        const int base = (k >> 1) * 256 + (k & 1);
        sW[base + (n + 0) * 2] = w.x;
        sW[base + (n + 1) * 2] = w.y;
        sW[base + (n + 2) * 2] = w.z;
        sW[base + (n + 3) * 2] = w.w;
    }
    __syncthreads();

    const int wave = tid >> 5;
    const int lane = tid & 31;
    const int row0 = blockIdx.x * 128 + wave * 16;
    if (row0 >= N) return;              // wave-uniform exit: EXEC stays all-1s

    const int halfsel = lane >> 4;      // 0: lanes 0-15, 1: lanes 16-31
    const int l15     = lane & 15;

    v8f acc[8];
#pragma unroll
    for (int t = 0; t < 8; ++t) acc[t] = (v8f)0.0f;

    // A fragment: lane<16 -> K = k0+0,k0+1 ; lane>=16 -> K = k0+2,k0+3 ; M = l15
    const float* Xrow = X + (size_t)(row0 + l15) * HID;
    const int kA = halfsel * 2;

    // prefetch back half of this lane's A row (global_prefetch_b8)
    __builtin_prefetch(Xrow + 64, 0, 1);

    for (int k0 = 0; k0 < HID; k0 += 4) {
        v2f a = *(const v2f*)(Xrow + k0 + kA);
        // pair index for this lane half: rows (k0+2h, k0+2h+1)
        const float* bbase = sW + ((k0 >> 1) + halfsel) * 256 + l15 * 2;
#pragma unroll
        for (int t = 0; t < 8; ++t) {
            v2f b = *(const v2f*)(bbase + t * 32);   // single aligned ds_load_b64
            acc[t] = __builtin_amdgcn_wmma_f32_16x16x4_f32(
                false, a, false, b, (short)0, acc[t], false, false);
        }
    }

    // C/D layout: VGPR j holds C[M = j + 8*halfsel][N = l15] per 16-col tile
    float* Hout = H + (size_t)row0 * HID;
#pragma unroll
    for (int t = 0; t < 8; ++t) {
#pragma unroll
        for (int j = 0; j < 8; ++j) {
            Hout[(size_t)(j + halfsel * 8) * HID + t * 16 + l15] = acc[t][j];
        }
    }
}

// ---------------------------------------------------------------------------
// AGG init with self-loop term: AGG = H * dis^2
// ---------------------------------------------------------------------------
__global__ void agg_init(float* __restrict__ AGG, const float* __restrict__ H,
                         const float* __restrict__ dis, int total) {
    int idx = blockIdx.x * blockDim.x + threadIdx.x;
    if (idx >= total) return;
    int row = idx >> 7;
    float d = dis[row];
    AGG[idx] = H[idx] * d * d;
}

// ---------------------------------------------------------------------------
// edge message scatter: AGG[dst] += H[src] * dis[src]*dis[dst]
// one wave (32 lanes) per edge, 4 channels (float4) per lane.
// ---------------------------------------------------------------------------
__global__ void edge_msg(const float* __restrict__ H, const float* __restrict__ dis,
                         const int* __restrict__ src, const int* __restrict__ dst,
                         float* __restrict__ AGG, int E) {
    unsigned int gid = blockIdx.x * blockDim.x + threadIdx.x;
    unsigned int e = gid >> 5;
    if (e >= (unsigned int)E) return;
    int c = (gid & 31) * 4;
    int s = src[e];
    int d = dst[e];
    float coef = dis[s] * dis[d];
    float4 m = *(const float4*)(H + (size_t)s * HID + c);
    float* out = AGG + (size_t)d * HID + c;
    unsafeAtomicAdd(out + 0, m.x * coef);
    unsafeAtomicAdd(out + 1, m.y * coef);
    unsafeAtomicAdd(out + 2, m.z * coef);
    unsafeAtomicAdd(out + 3, m.w * coef);
}

// ---------------------------------------------------------------------------
// BatchNorm pass 1: per-channel sums of t and t^2, where t = relu(AGG + b)
// block = 128 threads, channel = tid, rows strided over grid.
// ---------------------------------------------------------------------------
__global__ void bn_stats(const float* __restrict__ AGG, const float* __restrict__ b,
                         float* __restrict__ stats, int N) {
    int c = threadIdx.x;               // 0..127
    float bias = b[c];
    float s1 = 0.0f, s2 = 0.0f;
    for (int r = blockIdx.x; r < N; r += gridDim.x) {
        float t = AGG[(size_t)r * HID + c] + bias;
        t = fmaxf(t, 0.0f);
        s1 += t;
        s2 += t * t;
    }
    unsafeAtomicAdd(&stats[c], s1);
    unsafeAtomicAdd(&stats[HID + c], s2);
}

// BatchNorm pass 2: mean / invstd (biased variance, eps = 1e-5)
__global__ void bn_finalize(const float* __restrict__ stats, float* __restrict__ stats2, int N) {
    int c = threadIdx.x;
    float invN = 1.0f / (float)N;
    float m = stats[c] * invN;
    float v = stats[HID + c] * invN - m * m;
    stats2[c]       = m;
    stats2[HID + c] = rsqrtf(v + EPS);
}

// BatchNorm pass 3: OUT = g*(relu(AGG+b) - mean)*invstd + beta
__global__ void bn_normalize(const float* __restrict__ AGG, const float* __restrict__ b,
                             const float* __restrict__ stats2,
                             const float* __restrict__ g, const float* __restrict__ be,
                             float* __restrict__ OUT, int total) {
    int idx = blockIdx.x * blockDim.x + threadIdx.x;
    if (idx >= total) return;
    int c = idx & (HID - 1);
    float t = fmaxf(AGG[idx] + b[c], 0.0f);
    OUT[idx] = g[c] * (t - stats2[c]) * stats2[HID + c] + be[c];
}

// ---------------------------------------------------------------------------
// global max pool per graph (atomicMax on float via int punning; init = -inf)
// ---------------------------------------------------------------------------
__device__ __forceinline__ void atomicMaxF(float* addr, float v) {
    if (v >= 0.0f) atomicMax((int*)addr, __float_as_int(v));
    else           atomicMin((unsigned int*)addr, __float_as_uint(v));
}

__global__ void pool_max(const float* __restrict__ X, const int* __restrict__ batch,
                         float* __restrict__ pooled, int total) {
    int idx = blockIdx.x * blockDim.x + threadIdx.x;
    if (idx >= total) return;
    int row = idx >> 7;
    int c   = idx & (HID - 1);
    atomicMaxF(&pooled[batch[row] * HID + c], X[idx]);
}

// ---------------------------------------------------------------------------
// MLP head: out[g] = relu(pooled[g] @ lw1 + lb1) @ lw2 + lb2
// one block per graph, 64 threads (one per hidden unit)
// ---------------------------------------------------------------------------
__global__ void mlp_head(const float* __restrict__ pooled,
                         const float* __restrict__ lw1, const float* __restrict__ lb1,
                         const float* __restrict__ lw2, const float* __restrict__ lb2,
                         float* __restrict__ out) {
    __shared__ float sh[64];
    int g = blockIdx.x;
    int j = threadIdx.x;          // 0..63
    float acc = lb1[j];
    const float* p = pooled + g * HID;
    for (int k = 0; k < HID; ++k) acc += p[k] * lw1[k * 64 + j];
    sh[j] = fmaxf(acc, 0.0f);
    __syncthreads();
    if (j == 0) {
        float o = lb2[0];
        for (int k = 0; k < 64; ++k) o += sh[k] * lw2[k];
        out[g] = o;
    }
}

// ---------------------------------------------------------------------------
// launcher
// ---------------------------------------------------------------------------
extern "C" void kernel_launch(void* const* d_in, const int* in_sizes, int n_in,
                              void* d_out, int out_size, void* d_ws, size_t ws_size,
                              hipStream_t stream) {
    const float* x    = (const float*)d_in[0];
    const int*   eidx = (const int*)d_in[2];
    const int*   batch = (const int*)d_in[3];
    const float* W[3]  = { (const float*)d_in[4],  (const float*)d_in[8],  (const float*)d_in[12] };
    const float* bb[3] = { (const float*)d_in[5],  (const float*)d_in[9],  (const float*)d_in[13] };
    const float* gg[3] = { (const float*)d_in[6],  (const float*)d_in[10], (const float*)d_in[14] };
    const float* be[3] = { (const float*)d_in[7],  (const float*)d_in[11], (const float*)d_in[15] };
    const float* lw1 = (const float*)d_in[16];
    const float* lb1 = (const float*)d_in[17];
    const float* lw2 = (const float*)d_in[18];
    const float* lb2 = (const float*)d_in[19];
    float* out = (float*)d_out;

    const int N = in_sizes[0] / HID;
    const int E = in_sizes[2] / 2;
    const int* src = eidx;
    const int* dst = eidx + E;

    // workspace layout
    float* ws   = (float*)d_ws;
    size_t nf   = (size_t)N * HID;
    float* B0   = ws;                 // GEMM output H
    float* B1   = ws + nf;            // AGG
    float* B2   = ws + 2 * nf;        // normalized layer output
    float* dis  = ws + 3 * nf;        // [N]
    float* stats  = dis + N;          // [256] (sum, sumsq)
    float* stats2 = stats + 256;      // [256] (mean, invstd)
    float* pooled = stats2 + 256;     // [64*128]

    const int total = N * HID;
    const dim3 blk256(256), blk128(128), blk64(64);

    // degree -> dis = rsqrt(deg+1), computed once
    zero_f32<<<(N + 255) / 256, blk256, 0, stream>>>(dis, N);
    deg_accum<<<(E + 255) / 256, blk256, 0, stream>>>(dst, dis, E);
    deg_to_dis<<<(N + 255) / 256, blk256, 0, stream>>>(dis, N);

    const float* layer_in = x;
    for (int l = 0; l < 3; ++l) {
        // H = X @ W   (WMMA fp32)
        gemm_wmma_f32<<<(N + 127) / 128, blk256, 0, stream>>>(layer_in, W[l], B0, N);
        // AGG = H * dis^2  (self loop)
        agg_init<<<(total + 255) / 256, blk256, 0, stream>>>(B1, B0, dis, total);
        // AGG[dst] += H[src] * dis[src]*dis[dst]
        {
            unsigned int threads = (unsigned int)E * 32u;
            edge_msg<<<(threads + 255u) / 256u, blk256, 0, stream>>>(B0, dis, src, dst, B1, E);
        }
        // BatchNorm(relu(AGG + b))
        zero_f32<<<1, blk256, 0, stream>>>(stats, 256);
        bn_stats<<<256, blk128, 0, stream>>>(B1, bb[l], stats, N);
        bn_finalize<<<1, blk128, 0, stream>>>(stats, stats2, N);
        bn_normalize<<<(total + 255) / 256, blk256, 0, stream>>>(B1, bb[l], stats2, gg[l], be[l], B2, total);
        layer_in = B2;
    }

    // global max pool per graph
    fill_neginf<<<(NUM_GRAPHS * HID + 255) / 256, blk256, 0, stream>>>(pooled, NUM_GRAPHS * HID);
    pool_max<<<(total + 255) / 256, blk256, 0, stream>>>(B2, batch, pooled, total);

    // MLP head
    mlp_head<<<NUM_GRAPHS, blk64, 0, stream>>>(pooled, lw1, lb1, lw2, lb2, out);
}